// STST_Block_17746804867253
// MI455X (gfx1250) — compile-verified
//
#include <hip/hip_runtime.h>
#include <hip/hip_bf16.h>

typedef float v2f __attribute__((ext_vector_type(2)));
typedef float v8f __attribute__((ext_vector_type(8)));
typedef int   v4i __attribute__((ext_vector_type(4)));

#define NB  64
#define CCH 128
#define TT  64
#define VV  25
#define SSP 3
#define LL  (TT*VV)   // 1600
#define EPSB 1e-5f
#define KCHUNK 256

__device__ __forceinline__ v8f wmma4(v2f a, v2f b, v8f c) {
  // D = A(16x4 f32) * B(4x16 f32) + C(16x16 f32)
  return __builtin_amdgcn_wmma_f32_16x16x4_f32(false, a, false, b, (short)0, c,
                                               false, false);
}

__device__ __forceinline__ v8f v8zero() {
  v8f z = {0.f,0.f,0.f,0.f,0.f,0.f,0.f,0.f};
  return z;
}

// 16-byte global -> LDS copy. Prefers the gfx1250 async-to-LDS path
// (ASYNCcnt-tracked, bypasses VGPRs); falls back to a sync vector copy.
__device__ __forceinline__ void cp_async16(float* l, const float* g) {
#if __has_builtin(__builtin_amdgcn_global_load_async_to_lds_b128)
  __builtin_amdgcn_global_load_async_to_lds_b128(
      (__attribute__((address_space(1))) v4i*)g,
      (__attribute__((address_space(3))) v4i*)l, 0, 0);
#else
  *(float4*)l = *(const float4*)g;
#endif
}

__device__ __forceinline__ void async_wait0() {
#if __has_builtin(__builtin_amdgcn_s_wait_asynccnt)
  __builtin_amdgcn_s_wait_asynccnt(0);
#else
  asm volatile("s_wait_asynccnt 0x0" ::: "memory");
#endif
}

// ---------------------------------------------------------------------------
// Generic per-sample GEMM: out[n] = W[OxK] @ X[n][KxL]  (+bias, +BN, +res, lrelu)
// Block: 256 threads = 8 waves covering a 128(M) x 32(N) output block.
// Each wave: one M tile x two N tiles (A fragment reused for 2 WMMAs).
// B slab [Kchunk x 32] staged in LDS via async-to-LDS, shared by all 8 waves.
// Grid: (L/32, N, ceil(O/128))
// ---------------------------------------------------------------------------
__global__ __launch_bounds__(256)
void k_gemm(const float* __restrict__ X, const float* __restrict__ W,
            const float* __restrict__ bias,
            const float* __restrict__ bg, const float* __restrict__ bb,
            const float* __restrict__ bm, const float* __restrict__ bv,
            const float* __restrict__ res,
            float* __restrict__ out,
            int K, int L, int O, int lrelu) {
  __shared__ float bs[KCHUNK * 32];
  const int tid  = threadIdx.x;
  const int lane = tid & 31;
  const int wave = tid >> 5;
  const int col0 = blockIdx.x * 32;
  const int n    = blockIdx.y;
  const int row0 = blockIdx.z * 128 + wave * 16;

  const float* Xn = X + (size_t)n * K * L;
  const int mr   = lane & 15;
  const int koff = (lane >= 16) ? 2 : 0;
  const int nc   = lane & 15;
  v8f acc0 = v8zero(), acc1 = v8zero();

  if (row0 < O) __builtin_prefetch(W + (size_t)(row0 + mr) * K, 0, 0);

  for (int kc = 0; kc < K; kc += KCHUNK) {
    const int kl = (K - kc < KCHUNK) ? (K - kc) : KCHUNK;
    if (kc) __syncthreads();
    const int nel = kl * 32;
    for (int e = tid * 4; e < nel; e += 1024) {
      int k = e >> 5, c = e & 31;
      cp_async16(&bs[e], &Xn[(size_t)(kc + k) * L + col0 + c]);
    }
    async_wait0();
    __syncthreads();

    if (row0 < O) {
      const float* Wr = W + (size_t)(row0 + mr) * K + kc;
#pragma unroll 8
      for (int kk = 0; kk < kl; kk += 4) {
        int kb = kk + koff;
        v2f a = *(const v2f*)(Wr + kb);
        v2f b0, b1;
        b0.x = bs[kb * 32 + nc];        b0.y = bs[(kb + 1) * 32 + nc];
        b1.x = bs[kb * 32 + 16 + nc];   b1.y = bs[(kb + 1) * 32 + 16 + nc];
        acc0 = wmma4(a, b0, acc0);
        acc1 = wmma4(a, b1, acc1);
      }
    }
  }

  if (row0 < O) {
    const int col   = col0 + nc;
    const int rbase = row0 + ((lane >= 16) ? 8 : 0);
    float* On = out + (size_t)n * O * L;
    const float* Rn = res ? (res + (size_t)n * O * L) : nullptr;
#pragma unroll
    for (int r = 0; r < 8; ++r) {
      int row = rbase + r;
      float sc = 1.f, sh = 0.f;
      if (bg) {
        sc = bg[row] * rsqrtf(bv[row] + EPSB);
        sh = bb[row] - bm[row] * sc;
      }
      float v0 = acc0[r] + bias[row];
      float v1 = acc1[r] + bias[row];
      if (bg) { v0 = v0 * sc + sh; v1 = v1 * sc + sh; }
      if (Rn) { v0 += Rn[row * L + col]; v1 += Rn[row * L + col + 16]; }
      if (lrelu) {
        v0 = (v0 >= 0.f) ? v0 : 0.1f * v0;
        v1 = (v1 >= 0.f) ? v1 : 0.1f * v1;
      }
      On[row * L + col]      = v0;
      On[row * L + col + 16] = v1;
    }
  }
}

// ---------------------------------------------------------------------------
// Spatial attention: att[n,s,t,u,v] = att0[s,u,v] + tanh((q^T k)/32)*alpha[s]
// One wave per (n,s,t): 32x32 padded output (2x2 WMMA tiles), K = IC = 32.
// attW stored padded [n][s][t][32*32] with zeros outside 25x25.
// ---------------------------------------------------------------------------
__global__ __launch_bounds__(256)
void k_attS(const float* __restrict__ QK, const float* __restrict__ att0,
            const float* __restrict__ alphas, float* __restrict__ attW) {
  int tid = threadIdx.x, lane = tid & 31, wave = tid >> 5;
  int gw = blockIdx.x * 8 + wave;              // n*S*T waves
  int t = gw % TT; int s = (gw / TT) % SSP; int n = gw / (TT * SSP);
  const float* qb = QK + ((size_t)n * 192 + s * 32) * LL + t * VV;       // q[c][u]
  const float* kb = QK + ((size_t)n * 192 + 96 + s * 32) * LL + t * VV;  // k[c][v]
  const int mr = lane & 15, koff = (lane >= 16) ? 2 : 0;
  const int u0 = mr, u1 = mr + 16;
  v8f a00 = v8zero(), a01 = v8zero(), a10 = v8zero(), a11 = v8zero();
#pragma unroll
  for (int kk = 0; kk < 32; kk += 4) {
    int k0 = kk + koff;
    v2f A0, A1, B0, B1;
    A0.x = qb[k0 * LL + u0];                 A0.y = qb[(k0 + 1) * LL + u0];
    A1.x = (u1 < VV) ? qb[k0 * LL + u1] : 0.f;
    A1.y = (u1 < VV) ? qb[(k0 + 1) * LL + u1] : 0.f;
    B0.x = kb[k0 * LL + u0];                 B0.y = kb[(k0 + 1) * LL + u0];
    B1.x = (u1 < VV) ? kb[k0 * LL + u1] : 0.f;
    B1.y = (u1 < VV) ? kb[(k0 + 1) * LL + u1] : 0.f;
    a00 = wmma4(A0, B0, a00); a01 = wmma4(A0, B1, a01);
    a10 = wmma4(A1, B0, a10); a11 = wmma4(A1, B1, a11);
  }
  float alpha = alphas[s];
  const float* a0s = att0 + (size_t)s * VV * VV;
  float* ow = attW + (((size_t)n * SSP + s) * TT + t) * 1024;
  int vcol = lane & 15;
  int rb = (lane >= 16) ? 8 : 0;
  v8f* accs[4] = {&a00, &a01, &a10, &a11};
#pragma unroll
  for (int mi = 0; mi < 2; ++mi)
#pragma unroll
    for (int ni = 0; ni < 2; ++ni) {
      v8f acc = *accs[mi * 2 + ni];
#pragma unroll
      for (int r = 0; r < 8; ++r) {
        int u = mi * 16 + rb + r;
        int v = ni * 16 + vcol;
        float val = 0.f;
        if (u < VV && v < VV)
          val = a0s[u * VV + v] + tanhf(acc[r] * (1.f / 32.f)) * alpha;
        ow[u * 32 + v] = val;
      }
    }
}

// ---------------------------------------------------------------------------
// Apply spatial attention: y1[n, s*128+c, t, v] = sum_u x[n,c,t,u]*att[n,s,t,u,v]
// One block per (n,s,t): att tile (32x32) async-staged in LDS.
// ---------------------------------------------------------------------------
__global__ __launch_bounds__(256)
void k_applyS(const float* __restrict__ x, const float* __restrict__ attW,
              float* __restrict__ y1) {
  __shared__ float bs[1024];
  int tid = threadIdx.x, lane = tid & 31, wave = tid >> 5;
  int b = blockIdx.x;
  int t = b % TT; int s = (b / TT) % SSP; int n = b / (TT * SSP);
  const float* aw = attW + (((size_t)n * SSP + s) * TT + t) * 1024;
  cp_async16(&bs[tid * 4], &aw[tid * 4]);
  async_wait0();
  __syncthreads();
  const int mr = lane & 15, koff = (lane >= 16) ? 2 : 0, nc = lane & 15;
  const int row0 = wave * 16;
  const float* xr = x + ((size_t)n * CCH + row0 + mr) * LL + t * VV;  // A[m][u]
  v8f acc0 = v8zero(), acc1 = v8zero();
#pragma unroll
  for (int kk = 0; kk < 32; kk += 4) {
    int k0 = kk + koff;
    v2f a;
    a.x = (k0 < VV) ? xr[k0] : 0.f;
    a.y = (k0 + 1 < VV) ? xr[k0 + 1] : 0.f;
    v2f b0, b1;
    b0.x = bs[k0 * 32 + nc];        b0.y = bs[(k0 + 1) * 32 + nc];
    b1.x = bs[k0 * 32 + 16 + nc];   b1.y = bs[(k0 + 1) * 32 + 16 + nc];
    acc0 = wmma4(a, b0, acc0);
    acc1 = wmma4(a, b1, acc1);
  }
  float* yo = y1 + ((size_t)n * 384 + s * 128) * LL + t * VV;
  int rb = row0 + ((lane >= 16) ? 8 : 0);
#pragma unroll
  for (int r = 0; r < 8; ++r) {
    int row = rb + r;
    yo[row * LL + nc] = acc0[r];
    int v1 = 16 + nc;
    if (v1 < VV) yo[row * LL + v1] = acc1[r];
  }
}

// xbar[n,c,t] = mean_v s_out[n,c,t,v]   (fold mean before W_in_t GEMM)
__global__ void k_mean(const float* __restrict__ sout, float* __restrict__ xbar) {
  int i = blockIdx.x * 256 + threadIdx.x;
  if (i >= NB * CCH * TT) return;
  int t = i % TT; int c = (i / TT) % CCH; int n = i / (TT * CCH);
  const float* p = sout + ((size_t)n * CCH + c) * LL + t * VV;
  float sum = 0.f;
  for (int v = 0; v < VV; ++v) sum += p[v];
  xbar[i] = sum * (1.f / 25.f);
}

// ---------------------------------------------------------------------------
// Temporal attention: att[n,d,s,t,q] = mask * tanh((q^T k)/32) * alpha[d][s]
// One wave per (n,d,s, 16x16 tile of 64x64). K = IC = 32.
// ---------------------------------------------------------------------------
__global__ __launch_bounds__(256)
void k_attT(const float* __restrict__ QT, const float* __restrict__ alphaf,
            const float* __restrict__ alphab, float* __restrict__ attT) {
  int tid = threadIdx.x, lane = tid & 31, wave = tid >> 5;
  int gw = blockIdx.x * 8 + wave;
  int tile = gw & 15, combo = gw >> 4;
  int s = combo & 1, d = (combo >> 1) & 1, n = combo >> 2;
  int t0 = (tile >> 2) * 16, q0 = (tile & 3) * 16;
  const float* Qn = QT + (size_t)n * 256 * 64;
  const float* qb = Qn + (size_t)(d * 64 + s * 32) * 64;        // q[c][t]
  const float* kb = Qn + (size_t)(128 + d * 64 + s * 32) * 64;  // k[c][q]
  const int mr = lane & 15, koff = (lane >= 16) ? 2 : 0, nc = lane & 15;
  v8f acc = v8zero();
#pragma unroll
  for (int kk = 0; kk < 32; kk += 4) {
    int k0 = kk + koff;
    v2f a; a.x = qb[k0 * 64 + t0 + mr]; a.y = qb[(k0 + 1) * 64 + t0 + mr];
    v2f b; b.x = kb[k0 * 64 + q0 + nc]; b.y = kb[(k0 + 1) * 64 + q0 + nc];
    acc = wmma4(a, b, acc);
  }
  float alpha = (d == 0) ? alphaf[s] : alphab[s];
  float* ow = attT + ((size_t)n * 4 + d * 2 + s) * 4096;
  int rb = t0 + ((lane >= 16) ? 8 : 0);
  int q = q0 + nc;
#pragma unroll
  for (int r = 0; r < 8; ++r) {
    int t = rb + r;
    float val = tanhf(acc[r] * (1.f / 32.f)) * alpha;
    bool keep = (d == 0) ? (t >= q) : (q >= t);   // fmask / bmask
    ow[t * 64 + q] = keep ? val : 0.f;
  }
}

// ---------------------------------------------------------------------------
// Apply temporal attention: z[n, ds*128+c, q, v] = sum_t t_in[n,c,t,v]*att[t,q]
// Block per (n,ds,v): att (64x64) async-staged in LDS; 4 q-tiles per wave.
// ---------------------------------------------------------------------------
__global__ __launch_bounds__(256)
void k_applyT(const float* __restrict__ tin, const float* __restrict__ attT,
              float* __restrict__ z) {
  __shared__ float bs[4096];
  int tid = threadIdx.x, lane = tid & 31, wave = tid >> 5;
  int v = blockIdx.x; int ds = blockIdx.y; int n = blockIdx.z;
  const float* aw = attT + ((size_t)n * 4 + ds) * 4096;
#pragma unroll
  for (int e = tid * 4; e < 4096; e += 1024) cp_async16(&bs[e], &aw[e]);
  async_wait0();
  __syncthreads();
  const int mr = lane & 15, koff = (lane >= 16) ? 2 : 0, nc = lane & 15;
  const int row0 = wave * 16;
  const float* xr = tin + ((size_t)n * CCH + row0 + mr) * LL + v;  // A[c][t]
  v8f acc[4] = {v8zero(), v8zero(), v8zero(), v8zero()};
#pragma unroll 4
  for (int kk = 0; kk < 64; kk += 4) {
    int k0 = kk + koff;
    v2f a; a.x = xr[k0 * VV]; a.y = xr[(k0 + 1) * VV];
#pragma unroll
    for (int ni = 0; ni < 4; ++ni) {
      v2f b; b.x = bs[k0 * 64 + ni * 16 + nc]; b.y = bs[(k0 + 1) * 64 + ni * 16 + nc];
      acc[ni] = wmma4(a, b, acc[ni]);
    }
  }
  float* zo = z + ((size_t)n * 512 + ds * 128) * LL + v;  // ds*128 == d*256+s*128
  int rb = row0 + ((lane >= 16) ? 8 : 0);
#pragma unroll
  for (int ni = 0; ni < 4; ++ni) {
    int q = ni * 16 + nc;
#pragma unroll
    for (int r = 0; r < 8; ++r) {
      zo[(size_t)(rb + r) * LL + q * VV] = acc[ni][r];
    }
  }
}

// ---------------------------------------------------------------------------
// Temporal conv k=7 (pad 3) + bias + BN + residual + lrelu -> d_out
// 7 accumulated GEMMs with T-shifted B staging in LDS (scalar staging: the
// zero-padded boundary makes 16B async groups straddle validity).
// ---------------------------------------------------------------------------
__global__ __launch_bounds__(256)
void k_tcn(const float* __restrict__ tout, const float* __restrict__ Wt,
           const float* __restrict__ bias,
           const float* __restrict__ bg, const float* __restrict__ bb,
           const float* __restrict__ bm, const float* __restrict__ bv,
           float* __restrict__ out) {
  __shared__ float bs[128 * 16];
  int tid = threadIdx.x, lane = tid & 31, wave = tid >> 5;
  int col0 = blockIdx.x * 16; int n = blockIdx.y;
  const float* Xn = tout + (size_t)n * CCH * LL;
  const int mr = lane & 15, koff = (lane >= 16) ? 2 : 0, nc = lane & 15;
  const int row0 = wave * 16;
  v8f acc = v8zero();
  for (int dt = 0; dt < 7; ++dt) {
    int shift = (dt - 3) * VV;  // T shift moves column by multiples of V
    __syncthreads();
    for (int e = tid; e < 128 * 16; e += 256) {
      int k = e >> 4, c = e & 15;
      int col = col0 + c + shift;
      bs[e] = (col >= 0 && col < LL) ? Xn[k * LL + col] : 0.f;
    }
    __syncthreads();
    const float* Wr = Wt + (size_t)(row0 + mr) * CCH * 7 + dt;
#pragma unroll 4
    for (int kk = 0; kk < 128; kk += 4) {
      int k0 = kk + koff;
      v2f a; a.x = Wr[k0 * 7]; a.y = Wr[(k0 + 1) * 7];
      v2f b; b.x = bs[k0 * 16 + nc]; b.y = bs[(k0 + 1) * 16 + nc];
      acc = wmma4(a, b, acc);
    }
  }
  int col = col0 + nc;
  int rb = row0 + ((lane >= 16) ? 8 : 0);
  float* On = out + (size_t)n * CCH * LL;
#pragma unroll
  for (int r = 0; r < 8; ++r) {
    int row = rb + r;
    float val = acc[r] + bias[row];
    float sc = bg[row] * rsqrtf(bv[row] + EPSB);
    val = (val - bm[row]) * sc + bb[row];
    val += Xn[row * LL + col];
    val = (val >= 0.f) ? val : 0.1f * val;
    On[row * LL + col] = val;
  }
}

extern "C" void kernel_launch(void* const* d_in, const int* in_sizes, int n_in,
                              void* d_out, int out_size, void* d_ws, size_t ws_size,
                              hipStream_t stream) {
  const float* x       = (const float*)d_in[0];
  const float* W_in_s  = (const float*)d_in[1];
  const float* b_in_s  = (const float*)d_in[2];
  const float* att0s   = (const float*)d_in[3];
  const float* alphas  = (const float*)d_in[4];
  const float* W_out_s = (const float*)d_in[5];
  const float* b_out_s = (const float*)d_in[6];
  const float* g_out_s = (const float*)d_in[7];
  const float* be_out_s= (const float*)d_in[8];
  const float* m_out_s = (const float*)d_in[9];
  const float* v_out_s = (const float*)d_in[10];
  const float* W_ff_s  = (const float*)d_in[11];
  const float* b_ff_s  = (const float*)d_in[12];
  const float* g_ff_s  = (const float*)d_in[13];
  const float* be_ff_s = (const float*)d_in[14];
  const float* m_ff_s  = (const float*)d_in[15];
  const float* v_ff_s  = (const float*)d_in[16];
  const float* alphat_f= (const float*)d_in[17];
  const float* alphat_b= (const float*)d_in[18];
  const float* W_in_t  = (const float*)d_in[19];
  const float* b_in_t  = (const float*)d_in[20];
  const float* W_out_t = (const float*)d_in[21];
  const float* b_out_t = (const float*)d_in[22];
  const float* g_out_t = (const float*)d_in[23];
  const float* be_out_t= (const float*)d_in[24];
  const float* m_out_t = (const float*)d_in[25];
  const float* v_out_t = (const float*)d_in[26];
  const float* W_ff_t  = (const float*)d_in[27];
  const float* b_ff_t  = (const float*)d_in[28];
  const float* g_ff_t  = (const float*)d_in[29];
  const float* be_ff_t = (const float*)d_in[30];
  const float* m_ff_t  = (const float*)d_in[31];
  const float* v_ff_t  = (const float*)d_in[32];
  const float* W_tcn   = (const float*)d_in[33];
  const float* b_tcn   = (const float*)d_in[34];
  const float* g_tcn   = (const float*)d_in[35];
  const float* be_tcn  = (const float*)d_in[36];
  const float* m_tcn   = (const float*)d_in[37];
  const float* v_tcn   = (const float*)d_in[38];
  (void)in_sizes; (void)n_in; (void)out_size;

  // Workspace layout (floats), lifetime-based region reuse:
  //  RA (52,428,800): y1[n,384,1600] then z[n,512,1600]
  //  RB (19,660,800): QK_s[n,192,1600] then z1[n,128,1600]
  //  RC (12,582,912): attS padded, later att_t + xbar + QT
  //  RD (13,107,200): B1[n,128,1600] then t_out
  //  RE (13,107,200): s_out (long-lived)
  float* ws = (float*)d_ws;
  float* RA = ws;
  float* RB = RA + 52428800;
  float* RC = RB + 19660800;
  float* RD = RC + 12582912;
  float* RE = RD + 13107200;
  const size_t need_bytes = (size_t)(52428800 + 19660800 + 12582912 +
                                     13107200 + 13107200) * sizeof(float);
  if (ws_size < need_bytes) return;

  float* QK   = RB;                 // [n][192][1600]
  float* attS = RC;                 // [n][3][64][32*32]
  float* y1   = RA;                 // [n][384][1600]
  float* B1   = RD;                 // [n][128][1600]
  float* sout = RE;                 // [n][128][1600]  (t_in)
  float* attT_buf = RC;             // [n][4][64*64]
  float* xbar = RC + 1048576;       // [n][128][64]
  float* QT   = RC + 1572864;       // [n][256][64]
  float* z    = RA;                 // [n][512][1600]
  float* z1   = RB;                 // [n][128][1600]
  float* t_out= RD;                 // [n][128][1600]
  float* outp = (float*)d_out;

  dim3 blk(256);

  // 1) qk = W_in_s @ x + b         [192 x 128] @ [128 x 1600] per n
  k_gemm<<<dim3(50, 64, 2), blk, 0, stream>>>(
      x, W_in_s, b_in_s, nullptr, nullptr, nullptr, nullptr, nullptr,
      QK, 128, LL, 192, 0);

  // 2) spatial attention (tanh(q^T k / 32)*alpha + att0), padded 32x32
  k_attS<<<dim3(1536), blk, 0, stream>>>(QK, att0s, alphas, attS);

  // 3) y1 = x @ att per (n,s,t)
  k_applyS<<<dim3(NB * SSP * TT), blk, 0, stream>>>(x, attS, y1);

  // 4) B1 = lrelu(x + bn(W_out_s @ y1 + b))     K=384
  k_gemm<<<dim3(50, 64, 1), blk, 0, stream>>>(
      y1, W_out_s, b_out_s, g_out_s, be_out_s, m_out_s, v_out_s, x,
      B1, 384, LL, 128, 1);

  // 5) s_out = lrelu(x + bn(W_ff_s @ B1 + b))   K=128
  k_gemm<<<dim3(50, 64, 1), blk, 0, stream>>>(
      B1, W_ff_s, b_ff_s, g_ff_s, be_ff_s, m_ff_s, v_ff_s, x,
      sout, 128, LL, 128, 1);

  // 6) xbar = mean_v(s_out); QT = W_in_t @ xbar + b   (mean folded before GEMM)
  k_mean<<<dim3((NB * CCH * TT + 255) / 256), blk, 0, stream>>>(sout, xbar);
  k_gemm<<<dim3(2, 64, 2), blk, 0, stream>>>(
      xbar, W_in_t, b_in_t, nullptr, nullptr, nullptr, nullptr, nullptr,
      QT, 128, 64, 256, 0);

  // 7) temporal attention with causal masks
  k_attT<<<dim3(512), blk, 0, stream>>>(QT, alphat_f, alphat_b, attT_buf);

  // 8) z = t_in @ att_t per (n,dir,s,v)
  k_applyT<<<dim3(VV, 4, NB), blk, 0, stream>>>(sout, attT_buf, z);

  // 9) z1 = lrelu(t_in + bn(W_out_t @ z + b))   K=512
  k_gemm<<<dim3(50, 64, 1), blk, 0, stream>>>(
      z, W_out_t, b_out_t, g_out_t, be_out_t, m_out_t, v_out_t, sout,
      z1, 512, LL, 128, 1);

  // 10) t_out = lrelu(t_in + bn(W_ff_t @ z1 + b))  K=128
  k_gemm<<<dim3(50, 64, 1), blk, 0, stream>>>(
      z1, W_ff_t, b_ff_t, g_ff_t, be_ff_t, m_ff_t, v_ff_t, sout,
      t_out, 128, LL, 128, 1);

  // 11) out = lrelu(t_out + bn(tcn(t_out) + b))
  k_tcn<<<dim3(100, 64), blk, 0, stream>>>(
      t_out, W_tcn, b_tcn, g_tcn, be_tcn, m_tcn, v_tcn, outp);
}